// SNSCell_1082331759170
// MI455X (gfx1250) — compile-verified
//
#include <hip/hip_runtime.h>

typedef __attribute__((ext_vector_type(2))) float v2f;
typedef __attribute__((ext_vector_type(8))) float v8f;
typedef __attribute__((ext_vector_type(4))) float f4;

#define NN 128          // feature dim
#define ROWS 64         // batch rows per block
#define LDS_STRIDE 132  // 128 + 4 pad -> conflict-free column reads (132 % 64 == 4)

// ---------------------------------------------------------------------------
// Prep: clamp params, compute ge_col[j] = sum_i clamp01(Gmax[i,j])*clamp3(Esyn[i,j]),
// store clamped Gmax for the GEMM. One block per column j, one thread per row i.
// ---------------------------------------------------------------------------
__global__ __launch_bounds__(NN) void sns_prep(const float* __restrict__ Gm,
                                               const float* __restrict__ bm,
                                               const float* __restrict__ Gmax,
                                               const float* __restrict__ Esyn,
                                               float* __restrict__ gmax_c,
                                               float* __restrict__ gecol,
                                               float* __restrict__ gm_c,
                                               float* __restrict__ bm_c) {
    __shared__ float red[NN];
    const int j = blockIdx.x;
    const int i = threadIdx.x;

    float g = Gmax[i * NN + j];
    g = fminf(fmaxf(g, 0.0f), 1.0f);
    float e = Esyn[i * NN + j];
    e = fminf(fmaxf(e, -3.0f), 3.0f);

    gmax_c[i * NN + j] = g;
    red[i] = g * e;
    __syncthreads();
    #pragma unroll
    for (int s = NN / 2; s > 0; s >>= 1) {
        if (i < s) red[i] += red[i + s];
        __syncthreads();
    }
    if (i == 0) {
        gecol[j] = red[0];
        gm_c[j]  = fminf(fmaxf(Gm[j], 0.01f), 1.0f);
        bm_c[j]  = fminf(fmaxf(bm[j], -1.0f), 1.0f);
    }
}

// ---------------------------------------------------------------------------
// Main: fused  out = h + (-Gm*h + bm + i_app + clamp01(h)*(ge_col - h@Gmax))
// 8 waves/block; wave w owns column tile [16w,16w+16); block owns 64 batch rows.
// GEMM via v_wmma_f32_16x16x4_f32, K=128 -> 32 chained WMMAs per 16x16 tile.
// ---------------------------------------------------------------------------
__global__ __launch_bounds__(256) void sns_main(const float* __restrict__ i_app,
                                                const float* __restrict__ hidden,
                                                const float* __restrict__ gmax_c,
                                                const float* __restrict__ gecol,
                                                const float* __restrict__ gm_c,
                                                const float* __restrict__ bm_c,
                                                float* __restrict__ out,
                                                long long BN) {
    __shared__ float lds[ROWS * LDS_STRIDE];

    const int t    = threadIdx.x;
    const int lane = t & 31;
    const int w    = t >> 5;                      // wave id == column tile
    const long long rowBase = (long long)blockIdx.x * ROWS;

    // ---- stage hidden tile (64 x 128 fp32) into LDS, fully coalesced b128 ----
    {
        const f4* __restrict__ src = (const f4*)(hidden + rowBase * NN);
        #pragma unroll
        for (int it = 0; it < (ROWS * NN / 4) / 256; ++it) {   // 8 iters
            const int idx = t + it * 256;                       // float4 index
            const f4 v = src[idx];
            const int row = (idx * 4) >> 7;
            const int col = (idx * 4) & (NN - 1);
            float* dst = &lds[row * LDS_STRIDE + col];
            dst[0] = v.x; dst[1] = v.y; dst[2] = v.z; dst[3] = v.w;
        }
    }

    // ---- per-wave column-tile state ----
    const int nloc = lane & 15;
    const int jcol = w * 16 + nloc;               // global column this lane owns in B/C/D
    const int krow = (lane >> 4) * 2;             // K sub-offset per ISA A/B f32 layout

    // Preload all 32 B fragments (Gmax columns) -- 64 VGPRs, read from L2-hot ws.
    v2f bf[32];
    #pragma unroll
    for (int s = 0; s < 32; ++s) {
        const int k = 4 * s + krow;
        bf[s].x = gmax_c[k * NN + jcol];
        bf[s].y = gmax_c[(k + 1) * NN + jcol];
    }
    const float gm  = gm_c[jcol];
    const float bmv = bm_c[jcol];
    const float ge  = gecol[jcol];

    __syncthreads();

    #pragma unroll
    for (int mt = 0; mt < ROWS / 16; ++mt) {
        v8f acc = {0.f, 0.f, 0.f, 0.f, 0.f, 0.f, 0.f, 0.f};
        const float* __restrict__ arow = &lds[(mt * 16 + nloc) * LDS_STRIDE];

        #pragma unroll
        for (int s = 0; s < 32; ++s) {
            const int k = 4 * s + krow;
            v2f a;
            a.x = arow[k];
            a.y = arow[k + 1];
            // D = A(16x4) x B(4x16) + C  in fp32 (exact vs reference numerics)
            acc = __builtin_amdgcn_wmma_f32_16x16x4_f32(
                false, a, false, bf[s], (short)0, acc, false, false);
        }

        // ---- fused epilogue straight out of the C/D fragment layout ----
        #pragma unroll
        for (int r = 0; r < 8; ++r) {
            const int row = mt * 16 + r + 8 * (lane >> 4);    // M of acc[r] for this lane
            const long long b = rowBase + row;
            const float h   = lds[row * LDS_STRIDE + jcol];
            const float ia  = i_app[b * NN + jcol];
            const float hcl = fminf(fmaxf(h, 0.0f), 1.0f);
            const float val = h - gm * h + bmv + ia + hcl * (ge - acc[r]);
            const long long o = b * NN + jcol;
            out[o]      = val;   // tuple element 0
            out[BN + o] = val;   // tuple element 1 (same tensor returned twice)
        }
    }
}

extern "C" void kernel_launch(void* const* d_in, const int* in_sizes, int n_in,
                              void* d_out, int out_size, void* d_ws, size_t ws_size,
                              hipStream_t stream) {
    const float* i_app  = (const float*)d_in[0];
    const float* hidden = (const float*)d_in[1];
    const float* Gm     = (const float*)d_in[2];
    const float* bm     = (const float*)d_in[3];
    const float* Gmax   = (const float*)d_in[4];
    const float* Esyn   = (const float*)d_in[5];
    float* out = (float*)d_out;

    float* ws      = (float*)d_ws;
    float* gmax_c  = ws;                 // 128*128 floats
    float* gecol   = ws + NN * NN;       // 128
    float* gm_c    = gecol + NN;         // 128
    float* bm_c    = gm_c + NN;          // 128

    const long long BN    = (long long)in_sizes[0];   // B*N = 33,554,432
    const long long Brows = BN / NN;                  // 262144

    sns_prep<<<NN, NN, 0, stream>>>(Gm, bm, Gmax, Esyn, gmax_c, gecol, gm_c, bm_c);
    sns_main<<<(int)(Brows / ROWS), 256, 0, stream>>>(i_app, hidden, gmax_c, gecol,
                                                      gm_c, bm_c, out, BN);
}